// LinearPotential_74182675136843
// MI455X (gfx1250) — compile-verified
//
#include <hip/hip_runtime.h>
#include <hip/hip_bf16.h>

// ---------------------------------------------------------------------------
// out[i] = sum_j coeff[j] * exp(-||x_i - a_j||^2 / (2 w_j^2))
//
// Strategy (MI455X / gfx1250, wave32):
//  * sq_dist tile (16 rows x 16 anchors) via ONE v_wmma_f32_16x16x4_f32:
//      A row i  = [x, y, z, 1]                      (16x4, K padded to 4)
//      B col j  = [-2ax, -2ay, -2az, ||a||^2]       (4x16)
//      C init   = ||x_i||^2 broadcast along the row (C VGPR v <-> row v/v+8)
//    => D = ||x||^2 + ||a||^2 - 2 x.a  exactly, in fp32 (matches reference).
//  * tail per element: t = D * s_j ; e = exp2(t) ; acc = fma(e, c_j, acc)
//    with s_j = -log2(e)/(2 w_j^2) precomputed  (v_exp_f32 is TRANS, co-executes
//    with the XDL WMMA pipe).
//  * 24KB of per-anchor prepped data staged in LDS once per block.
// ---------------------------------------------------------------------------

typedef __attribute__((ext_vector_type(2))) float v2f;
typedef __attribute__((ext_vector_type(8))) float v8f;

#define N_LOC    262144
#define N_ANCHOR 1024
#define TILES_M  (N_LOC / 16)     // 16384 row tiles
#define TILES_N  (N_ANCHOR / 16)  // 64 anchor tiles
#define WAVES_PER_BLOCK 8

// Workspace layout (floats): bx[1024] by[1024] bz[1024] bw[1024] s[1024] c[1024]
#define WS_FLOATS (6 * N_ANCHOR)

__global__ __launch_bounds__(256)
void lp_prep_anchors(const float* __restrict__ aloc,
                     const float* __restrict__ coeff,
                     const float* __restrict__ params,
                     float* __restrict__ ws) {
    int j = blockIdx.x * blockDim.x + threadIdx.x;
    if (j < N_ANCHOR) {
        float ax = aloc[3 * j + 0];
        float ay = aloc[3 * j + 1];
        float az = aloc[3 * j + 2];
        ws[0 * N_ANCHOR + j] = -2.0f * ax;
        ws[1 * N_ANCHOR + j] = -2.0f * ay;
        ws[2 * N_ANCHOR + j] = -2.0f * az;
        ws[3 * N_ANCHOR + j] = ax * ax + ay * ay + az * az;
        float w = params[j];
        // exp(-d * 1/(2w^2)) == exp2(d * (-log2(e)/(2w^2)))
        ws[4 * N_ANCHOR + j] = -1.4426950408889634f * 0.5f / (w * w);
        ws[5 * N_ANCHOR + j] = coeff[j];
    }
}

__global__ __launch_bounds__(256)
void lp_main(const float* __restrict__ loc,
             const float* __restrict__ ws,
             float* __restrict__ out) {
    __shared__ float sm[WS_FLOATS];  // 24 KB of 320 KB/WGP

    // Stage all per-anchor prepped data into LDS once per block.
    for (int i = threadIdx.x; i < WS_FLOATS; i += 256) sm[i] = ws[i];
    __syncthreads();

    const int lane = threadIdx.x & 31;
    const int wave = threadIdx.x >> 5;
    const int tile = blockIdx.x * WAVES_PER_BLOCK + wave;  // 16-row tile of locations
    const int col  = lane & 15;          // N index / row index within tile
    const bool hi  = lane >= 16;         // lanes 16..31 carry K=2,3 operand halves
    const int rbase = hi ? 8 : 0;        // C/D rows held by this half-wave

    // ---- A operand: 16x4 fp32. lane m<16: (K0,K1)=(x,y); lane m+16: (K2,K3)=(z,1)
    const int m = tile * 16 + col;
    const float x = loc[3 * m + 0];
    const float y = loc[3 * m + 1];
    const float z = loc[3 * m + 2];
    const float xsq = fmaf(x, x, fmaf(y, y, z * z));

    v2f A;
    A.x = hi ? z : x;
    A.y = hi ? 1.0f : y;

    // ---- C init: C VGPR v holds row (v + rbase); broadcast ||x_row||^2.
    // Row r's xsq lives in lane r (and r+16); bpermute gather, hoisted out of loop.
    v8f Cinit;
#pragma unroll
    for (int v = 0; v < 8; ++v) Cinit[v] = __shfl(xsq, rbase + v, 32);

    float acc[8];
#pragma unroll
    for (int v = 0; v < 8; ++v) acc[v] = 0.0f;

    for (int t = 0; t < TILES_N; ++t) {
        const int j = t * 16 + col;
        // B operand: 4x16 fp32. lanes 0..15: rows K=0,1 ; lanes 16..31: rows K=2,3
        v2f B;
        B.x = hi ? sm[2 * N_ANCHOR + j] : sm[0 * N_ANCHOR + j];
        B.y = hi ? sm[3 * N_ANCHOR + j] : sm[1 * N_ANCHOR + j];
        const float s_j = sm[4 * N_ANCHOR + j];  // -log2e/(2w^2) for column j
        const float c_j = sm[5 * N_ANCHOR + j];  // coeff for column j

        // D[r][n] = ||x_r||^2 + ||a_n||^2 - 2 x_r . a_n   (exact fp32 sq-dist tile)
        v8f D = __builtin_amdgcn_wmma_f32_16x16x4_f32(
            /*neg_a=*/false, A, /*neg_b=*/false, B,
            /*c_mod=*/(short)0, Cinit, /*reuse_a=*/false, /*reuse_b=*/false);

#pragma unroll
        for (int v = 0; v < 8; ++v) {
            float e = __builtin_amdgcn_exp2f(D[v] * s_j);  // v_exp_f32 (TRANS)
            acc[v] = fmaf(e, c_j, acc[v]);
        }
    }

    // Reduce over the 16 columns: lanes 0..15 share rows 0..7, lanes 16..31 rows 8..15.
#pragma unroll
    for (int mask = 1; mask < 16; mask <<= 1) {
#pragma unroll
        for (int v = 0; v < 8; ++v) acc[v] += __shfl_xor(acc[v], mask, 32);
    }

    if (col == 0) {
        const int row0 = tile * 16 + rbase;
#pragma unroll
        for (int v = 0; v < 8; ++v) out[row0 + v] = acc[v];
    }
}

extern "C" void kernel_launch(void* const* d_in, const int* in_sizes, int n_in,
                              void* d_out, int out_size, void* d_ws, size_t ws_size,
                              hipStream_t stream) {
    const float* locations3d        = (const float*)d_in[0];
    const float* anchor_locations3d = (const float*)d_in[1];
    const float* anchor_coeffs      = (const float*)d_in[2];
    const float* anchor_parameters  = (const float*)d_in[3];
    float* out = (float*)d_out;
    float* ws  = (float*)d_ws;  // needs 6*1024*4 = 24 KB

    lp_prep_anchors<<<(N_ANCHOR + 255) / 256, 256, 0, stream>>>(
        anchor_locations3d, anchor_coeffs, anchor_parameters, ws);

    lp_main<<<TILES_M / WAVES_PER_BLOCK, 256, 0, stream>>>(locations3d, ws, out);
}